// CenterLoss_59717225283874
// MI455X (gfx1250) — compile-verified
//
#include <hip/hip_runtime.h>

typedef __attribute__((ext_vector_type(2))) float v2f;
typedef __attribute__((ext_vector_type(8))) float v8f;

#define B_     128
#define D_     128
#define T_     2048
#define C_     64
#define TC     64            // t-chunk per iteration
#define TP     (TC + 4)      // 68: rows 16B-aligned for b128 async, stride 4 mod 64 banks -> conflict-free
#define NSPLIT 2             // WGs per batch row  => grid = 256
#define NCHUNK ((T_ / NSPLIT) / TC)   // 16 chunks per WG
#define FEATBUF (D_ * TP)    // floats per feature buffer
#define MAX_PART 16

// Workspace layout (floats):
//   ws[0]      : sum(feat^2) (float atomics)
//   ws[8..71]  : per-class counts (uint32 atomics)
//   ws[128+..] : nPart partial-M buffers, each C_*D_ floats

__global__ void zero_ws_kernel(float* __restrict__ ws, int n) {
    int i = blockIdx.x * blockDim.x + threadIdx.x;
    if (i < n) ws[i] = 0.0f;
}

__global__ void __launch_bounds__(256)
centerloss_main_kernel(const float* __restrict__ feature,
                       const int*   __restrict__ label,
                       float*       __restrict__ ws,
                       int nPart)
{
    __shared__ __align__(16) float s_feat[2 * FEATBUF];   // ~69.6 KB double buffer
    __shared__ int      s_lbl[2 * TC];
    __shared__ unsigned s_cnt[C_];

    const int tid  = threadIdx.x;
    const int wave = tid >> 5;
    const int lane = tid & 31;

    const int b     = blockIdx.x / NSPLIT;
    const int tbase = (blockIdx.x % NSPLIT) * (T_ / NSPLIT);

    if (tid < C_) s_cnt[tid] = 0u;

    const float* fbase = feature + (size_t)b * D_ * T_;
    const int*   lbase = label   + (size_t)b * T_;

    // Raw LDS byte offsets (generic LDS address low 32 bits == LDS offset)
    const unsigned featBase = (unsigned)(size_t)(void*)s_feat;
    const unsigned lblBase  = (unsigned)(size_t)(void*)s_lbl;

    // Staging geometry: lane covers 4 consecutive t of one d-row per b128 issue
    const int drow = tid >> 4;         // 0..15
    const int tcol = (tid & 15) * 4;   // 0,4,...,60
    const int li   = tid >> 2;         // label index when (tid&3)==0 -> 8 active lanes per wave

    // Issue one chunk's async DMA: 8x b128 + 1x b32 per wave => ASYNCcnt += 9 uniformly
    auto issue = [&](int chunk) {
        const int t0 = tbase + chunk * TC;
        const unsigned bufF = (unsigned)(chunk & 1) * (unsigned)(FEATBUF * 4);
        const unsigned bufL = (unsigned)(chunk & 1) * (unsigned)(TC * 4);
#pragma unroll
        for (int i = 0; i < 8; ++i) {
            const int d = i * 16 + drow;
            const unsigned long long gp =
                (unsigned long long)(size_t)(fbase + (size_t)d * T_ + (t0 + tcol));
            const unsigned lp = featBase + bufF + (unsigned)(d * TP + tcol) * 4u;
            asm volatile("global_load_async_to_lds_b128 %0, %1, off"
                         :: "v"(lp), "v"(gp) : "memory");
        }
        {
            const unsigned long long gp =
                (unsigned long long)(size_t)(lbase + t0 + li);
            const unsigned lp = lblBase + bufL + (unsigned)li * 4u;
            if ((tid & 3) == 0)    // every wave keeps >=1 active lane -> uniform counter bump
                asm volatile("global_load_async_to_lds_b32 %0, %1, off"
                             :: "v"(lp), "v"(gp) : "memory");
        }
    };

    // Per-wave WMMA accumulators: 4 class tiles x (16 classes x 16 d-cols)
    v8f acc0 = {}; v8f acc1 = {}; v8f acc2 = {}; v8f acc3 = {};
    float sumsq = 0.0f;

    const int d0   = wave * 16;
    const int dcol = lane & 15;
    const int half = lane >> 4;
    const int crow = lane & 15;

    issue(0);

    for (int c = 0; c < NCHUNK; ++c) {
        // Software pipeline: start next chunk's DMA, then wait only for current chunk
        if (c + 1 < NCHUNK) {
            issue(c + 1);
            asm volatile("s_wait_asynccnt 0x9" ::: "memory");   // chunk c complete (in-order)
        } else {
            asm volatile("s_wait_asynccnt 0x0" ::: "memory");
        }
        __syncthreads();

        const int fOff = (c & 1) * FEATBUF;
        const int lOff = (c & 1) * TC;

        // per-WG histogram from staged labels
        if (tid < TC) atomicAdd(&s_cnt[s_lbl[lOff + tid] & (C_ - 1)], 1u);

        // one-hot GEMM: M_tile += Onehot(16xK) x Feat(Kx16), K-step 4
        const int rbase = fOff + (d0 + dcol) * TP;
#pragma unroll
        for (int k0 = 0; k0 < TC; k0 += 4) {
            const int kk = k0 + 2 * half;
            const int l0 = s_lbl[lOff + kk];
            const int l1 = s_lbl[lOff + kk + 1];
            v2f Bv;                               // B[k,d]: lane holds (K=kk+{0,1}, N=dcol)
            Bv.x = s_feat[rbase + kk];
            Bv.y = s_feat[rbase + kk + 1];
            sumsq += Bv.x * Bv.x + Bv.y * Bv.y;   // each element visited exactly once

            v2f Av;                               // A[m,k]: lane holds (M=crow, K=kk+{0,1})
            Av.x = (l0 == crow)      ? 1.0f : 0.0f;
            Av.y = (l1 == crow)      ? 1.0f : 0.0f;
            acc0 = __builtin_amdgcn_wmma_f32_16x16x4_f32(false, Av, false, Bv, (short)0, acc0, false, false);
            Av.x = (l0 == crow + 16) ? 1.0f : 0.0f;
            Av.y = (l1 == crow + 16) ? 1.0f : 0.0f;
            acc1 = __builtin_amdgcn_wmma_f32_16x16x4_f32(false, Av, false, Bv, (short)0, acc1, false, false);
            Av.x = (l0 == crow + 32) ? 1.0f : 0.0f;
            Av.y = (l1 == crow + 32) ? 1.0f : 0.0f;
            acc2 = __builtin_amdgcn_wmma_f32_16x16x4_f32(false, Av, false, Bv, (short)0, acc2, false, false);
            Av.x = (l0 == crow + 48) ? 1.0f : 0.0f;
            Av.y = (l1 == crow + 48) ? 1.0f : 0.0f;
            acc3 = __builtin_amdgcn_wmma_f32_16x16x4_f32(false, Av, false, Bv, (short)0, acc3, false, false);
        }
        __syncthreads();   // buffer (c&1) may be overwritten by issue(c+2) next iteration
    }

    // Fold accumulators into a partial-M buffer (contention spread over nPart buffers)
    float* M = ws + 128 + (size_t)(blockIdx.x % nPart) * (C_ * D_);
    const int dg = d0 + dcol;
#pragma unroll
    for (int j = 0; j < 8; ++j) {
        const int r = j + 8 * half;   // C/D layout: VGPR j -> row j (lanes 0-15) / row j+8 (lanes 16-31)
        atomicAdd(&M[(0 * 16 + r) * D_ + dg], acc0[j]);
        atomicAdd(&M[(1 * 16 + r) * D_ + dg], acc1[j]);
        atomicAdd(&M[(2 * 16 + r) * D_ + dg], acc2[j]);
        atomicAdd(&M[(3 * 16 + r) * D_ + dg], acc3[j]);
    }

    // sum(feat^2): wave32 shuffle reduce, one atomic per wave
    for (int off = 16; off > 0; off >>= 1)
        sumsq += __shfl_down(sumsq, off, 32);
    if (lane == 0) atomicAdd(&ws[0], sumsq);

    if (tid < C_) atomicAdd((unsigned*)(ws + 8) + tid, s_cnt[tid]);
}

__global__ void __launch_bounds__(256)
centerloss_final_kernel(const float* __restrict__ centers,
                        const float* __restrict__ ws,
                        int nPart,
                        float* __restrict__ out)
{
    __shared__ float s_part[256];
    const int tid = threadIdx.x;
    const unsigned* counts = (const unsigned*)(ws + 8);

    float partial = 0.0f;
    for (int idx = tid; idx < C_ * D_; idx += 256) {
        const int c = idx >> 7;
        float m = 0.0f;
        for (int p = 0; p < nPart; ++p) m += ws[128 + (size_t)p * (C_ * D_) + idx];
        const float cen   = centers[idx];
        const float cnt   = (float)counts[c];
        const float denom = fmaxf(cnt, 1.0f);
        out[1 + idx] = (cnt * cen - m) / denom;           // difference[c,d]
        partial += cnt * cen * cen - 2.0f * m * cen;      // loss cross terms
    }
    s_part[tid] = partial;
    __syncthreads();
    for (int s = 128; s > 0; s >>= 1) {
        if (tid < s) s_part[tid] += s_part[tid + s];
        __syncthreads();
    }
    if (tid == 0) {
        const float nd = (float)((size_t)B_ * T_) * (float)D_;
        out[0] = (ws[0] + s_part[0]) / nd;               // center_loss
    }
}

extern "C" void kernel_launch(void* const* d_in, const int* in_sizes, int n_in,
                              void* d_out, int out_size, void* d_ws, size_t ws_size,
                              hipStream_t stream) {
    const float* feature = (const float*)d_in[0];   // [B, D, T] f32
    const int*   label   = (const int*)d_in[1];     // [B, T] int
    const float* centers = (const float*)d_in[2];   // [C, D] f32
    float* out = (float*)d_out;                     // [1 + C*D]
    float* ws  = (float*)d_ws;

    const size_t headBytes = 128 * sizeof(float);
    const size_t partBytes = (size_t)C_ * D_ * sizeof(float);
    int nPart = 1;
    if (ws_size > headBytes + partBytes) {
        size_t m = (ws_size - headBytes) / partBytes;
        nPart = (m > MAX_PART) ? MAX_PART : (int)m;
        if (nPart < 1) nPart = 1;
    }

    const int zeroN = 128 + nPart * C_ * D_;
    zero_ws_kernel<<<(zeroN + 255) / 256, 256, 0, stream>>>(ws, zeroN);
    centerloss_main_kernel<<<B_ * NSPLIT, 256, 0, stream>>>(feature, label, ws, nPart);
    centerloss_final_kernel<<<1, 256, 0, stream>>>(centers, ws, nPart, out);
}